// ATSSAssigner_47631187312972
// MI455X (gfx1250) — compile-verified
//
#include <hip/hip_runtime.h>
#include <stdint.h>

// ATSS assigner for MI455X (gfx1250, wave32).
// Phase 0: workspace init + per-anchor WMMA B-fragment precompute.
// Phase A: 4-wave workgroup per (batch, 16-gt group); each wave owns 1/4 of
//          each pyramid level; squared center distances via
//          V_WMMA_F32_16X16X4_F32; intra-wave LDS transpose with wave_barrier
//          only (no s_barrier in the hot loop); 8-way top-9 merge per level;
//          mean+std threshold; sparse positive scatter via atomics.
// Phase B: per-(batch, anchor) resolution + output emission (store-roofline).

#define BATCH   32
#define NGT     128
#define NANCH   8400
#define NTOPK   9
#define NCAND   27            // 3 levels * 9
#define NCLS    80
#define EPSF    1e-9f
#define NWAVES  4

typedef __attribute__((ext_vector_type(2))) float v2f;
typedef __attribute__((ext_vector_type(8))) float v8f;

__device__ __forceinline__ float iou_box(const float4 a, const float4 b) {
    float lx = fmaxf(a.x, b.x), ly = fmaxf(a.y, b.y);
    float rx = fminf(a.z, b.z), ry = fminf(a.w, b.w);
    float w  = fmaxf(rx - lx, 0.0f), h = fmaxf(ry - ly, 0.0f);
    float inter = w * h;
    float a1 = (a.z - a.x) * (a.w - a.y);
    float a2 = (b.z - b.x) * (b.w - b.y);
    return inter / (a1 + a2 - inter + EPSF);
}

// Zero the scatter workspace and precompute per-anchor WMMA B-fragments:
// B column for anchor a is [-2*acx, -2*acy, 1, acx^2+acy^2]^T.
__global__ void init_prep_kernel(const float* __restrict__ anchors,
                                 int* __restrict__ cnt,
                                 unsigned long long* __restrict__ best,
                                 float2* __restrict__ bf0,
                                 float2* __restrict__ bf1) {
    int i = blockIdx.x * blockDim.x + threadIdx.x;
    if (i < BATCH * NANCH) { cnt[i] = 0; best[i] = 0ull; }
    if (i < NANCH) {
        float4 ab = ((const float4*)anchors)[i];
        float acx = (ab.x + ab.z) * 0.5f, acy = (ab.y + ab.w) * 0.5f;
        bf0[i] = make_float2(-2.0f * acx, -2.0f * acy);
        bf1[i] = make_float2(1.0f, acx * acx + acy * acy);
    }
}

__global__ __launch_bounds__(NWAVES * 32)
void atss_phaseA_kernel(const float* __restrict__ anchors,
                        const float* __restrict__ gt_bboxes,
                        const float* __restrict__ pad_mask,
                        const float2* __restrict__ bf0,
                        const float2* __restrict__ bf1,
                        int* __restrict__ cnt,
                        unsigned long long* __restrict__ best) {
    __shared__ float  tile[NWAVES][16][17];     // wave-private transpose staging
    __shared__ float4 gtb_s[16];
    __shared__ int    valid_s[16];
    __shared__ int    candIdx[16][NCAND];
    __shared__ float  candIou[16][NCAND];
    __shared__ float  thr_s[16];
    __shared__ float  mergeD[NWAVES * 32][NTOPK];
    __shared__ int    mergeI[NWAVES * 32][NTOPK];

    const int tid  = threadIdx.x;
    const int lane = tid & 31;
    const int wid  = tid >> 5;
    const int half = lane >> 4;          // 0: K0/K1 fragment half, 1: K2/K3
    const int q    = lane & 15;          // gt row (A) / anchor col (B)
    const int b    = blockIdx.x >> 3;    // one (b, 16-gt group) per block
    const int g0   = (blockIdx.x & 7) << 4;

    const float4* __restrict__ anch4 = (const float4*)anchors;
    const float4* __restrict__ gtb4  = (const float4*)gt_bboxes;

    // Stage gt boxes + validity; build loop-invariant A fragment (same for all waves).
    float4 mygt = gtb4[b * NGT + g0 + q];
    if (tid < 16) {
        gtb_s[tid]   = gtb4[b * NGT + g0 + tid];
        valid_s[tid] = (pad_mask[b * NGT + g0 + tid] > 0.0f) ? 1 : 0;
    }
    float gcx = (mygt.x + mygt.z) * 0.5f;
    float gcy = (mygt.y + mygt.w) * 0.5f;
    // A row m = [gx, gy, gx^2+gy^2, 1]; lanes 0-15 hold K0,K1; lanes 16-31 K2,K3.
    v2f afrag;
    if (half == 0) { afrag.x = gcx;                   afrag.y = gcy;  }
    else           { afrag.x = gcx * gcx + gcy * gcy; afrag.y = 1.0f; }
    const float2* __restrict__ bsrc = (half == 0) ? bf0 : bf1;
    __syncthreads();

    const int lvl_start[3] = {0, 6400, 8000};
    const int lvl_len[3]   = {6400, 1600, 400};

    for (int lv = 0; lv < 3; ++lv) {
        float td[NTOPK]; int ti[NTOPK];
        #pragma unroll
        for (int j = 0; j < NTOPK; ++j) { td[j] = 3.0e38f; ti[j] = -1; }
        float worst = 3.0e38f; int wslot = 0;

        // Each wave owns a contiguous quarter of this level's anchors.
        const int seg    = lvl_len[lv] >> 2;
        const int mybase = lvl_start[lv] + wid * seg;
        for (int base = mybase; base < mybase + seg; base += 16) {
            float2 bv = bsrc[base + q];
            int pf = base + q + 16; if (pf >= NANCH) pf = NANCH - 1;
            __builtin_prefetch(&bsrc[pf], 0, 0);            // global_prefetch_b8
            v2f bfrag; bfrag.x = bv.x; bfrag.y = bv.y;
            v8f cz = {0.f, 0.f, 0.f, 0.f, 0.f, 0.f, 0.f, 0.f};
            // D[m][n] = squared center distance gt m <-> anchor base+n
            v8f d2 = __builtin_amdgcn_wmma_f32_16x16x4_f32(
                false, afrag, false, bfrag, (short)0, cz, false, false);
            #pragma unroll
            for (int r = 0; r < 8; ++r) tile[wid][half * 8 + r][q] = d2[r];
            // Intra-wave exchange only: LDS is in-order per wave; just stop
            // the compiler from reordering (no s_barrier needed).
            __builtin_amdgcn_fence(__ATOMIC_ACQ_REL, "wavefront");
            __builtin_amdgcn_wave_barrier();
            #pragma unroll
            for (int j = 0; j < 8; ++j) {
                float d  = tile[wid][q][half * 8 + j];
                int  idx = base + half * 8 + j;
                if (d < worst) {
                    td[wslot] = d; ti[wslot] = idx;
                    worst = td[0]; wslot = 0;
                    #pragma unroll
                    for (int j2 = 1; j2 < NTOPK; ++j2)
                        if (td[j2] > worst) { worst = td[j2]; wslot = j2; }
                }
            }
            __builtin_amdgcn_wave_barrier();                // WAR vs next store
        }

        // 8-way merge per gt (2 half-lanes x 4 waves) -> level top-9 indices.
        #pragma unroll
        for (int j = 0; j < NTOPK; ++j) { mergeD[tid][j] = td[j]; mergeI[tid][j] = ti[j]; }
        __syncthreads();
        if (tid < 16) {
            float md[NTOPK]; int mi[NTOPK];
            #pragma unroll
            for (int j = 0; j < NTOPK; ++j) { md[j] = 3.0e38f; mi[j] = -1; }
            float w2 = 3.0e38f; int ws2 = 0;
            for (int src = 0; src < 8; ++src) {
                int st = (src >> 1) * 32 + tid + ((src & 1) << 4);
                #pragma unroll
                for (int j = 0; j < NTOPK; ++j) {
                    float d = mergeD[st][j]; int idx = mergeI[st][j];
                    if (d < w2) {
                        md[ws2] = d; mi[ws2] = idx;
                        w2 = md[0]; ws2 = 0;
                        for (int j2 = 1; j2 < NTOPK; ++j2)
                            if (md[j2] > w2) { w2 = md[j2]; ws2 = j2; }
                    }
                }
            }
            #pragma unroll
            for (int j = 0; j < NTOPK; ++j) candIdx[tid][lv * NTOPK + j] = mi[j];
        }
        __syncthreads();                                    // WAR on mergeD/I
    }

    // Candidate IoUs: 16 gts x 27 candidates = 432 tasks across 128 threads.
    for (int t = tid; t < 16 * NCAND; t += NWAVES * 32) {
        int g = t / NCAND, k = t - g * NCAND;
        int idx = candIdx[g][k];
        candIou[g][k] = iou_box(gtb_s[g], anch4[idx]);
    }
    __syncthreads();
    // Per-gt threshold: mean + sample-std (ddof=1) over 27 candidate IoUs.
    if (tid < 16) {
        float s = 0.0f, ss = 0.0f;
        #pragma unroll
        for (int k = 0; k < NCAND; ++k) { float v = candIou[tid][k]; s += v; ss += v * v; }
        float mean = s * (1.0f / NCAND);
        float var  = (ss - s * mean) * (1.0f / (NCAND - 1));
        thr_s[tid] = mean + sqrtf(fmaxf(var, 0.0f));
    }
    __syncthreads();
    // Scatter positives: iou > thr AND anchor center strictly inside gt AND valid gt.
    for (int t = tid; t < 16 * NCAND; t += NWAVES * 32) {
        int g = t / NCAND, k = t - g * NCAND;
        if (!valid_s[g]) continue;
        float v = candIou[g][k];
        if (v <= thr_s[g]) continue;
        int idx = candIdx[g][k];
        float4 ab = anch4[idx];
        float acx = (ab.x + ab.z) * 0.5f, acy = (ab.y + ab.w) * 0.5f;
        float4 gb = gtb_s[g];
        float m = fminf(fminf(acx - gb.x, acy - gb.y), fminf(gb.z - acx, gb.w - acy));
        if (m > EPSF) {
            int slot = b * NANCH + idx;
            atomicAdd(&cnt[slot], 1);
            unsigned long long key =
                ((unsigned long long)__float_as_uint(v) << 32) | (unsigned)(g0 + g);
            atomicMax(&best[slot], key);
        }
    }
}

__global__ __launch_bounds__(256)
void atss_resolve_kernel(const float* __restrict__ anchors,
                         const int* __restrict__ gt_labels,
                         const float* __restrict__ gt_bboxes,
                         const float* __restrict__ pred_bboxes,
                         const int* __restrict__ bg_p,
                         const int* __restrict__ cnt,
                         const unsigned long long* __restrict__ best,
                         int* __restrict__ out_labels,
                         float* __restrict__ out_bboxes,
                         float* __restrict__ out_scores) {
    int tid = blockIdx.x * blockDim.x + threadIdx.x;
    if (tid >= BATCH * NANCH) return;
    int b = tid / NANCH, a = tid - b * NANCH;

    const float4* __restrict__ anch4 = (const float4*)anchors;
    const float4* __restrict__ gtb4  = (const float4*)gt_bboxes + (size_t)b * NGT;
    const float4* __restrict__ pred4 = (const float4*)pred_bboxes;

    int c = cnt[tid];
    int g = 0;
    if (c == 1) {
        g = (int)(best[tid] & 0xffffffffull);
    } else if (c > 1) {
        // Anchor matched multiple gts: argmax IoU over ALL gts
        // (reference includes padded gts here too).
        float4 ab = anch4[a];
        float bi = -1.0f;
        for (int j = 0; j < NGT; ++j) {
            float v = iou_box(gtb4[j], ab);
            if (v > bi) { bi = v; g = j; }
        }
    }
    bool pos = (c > 0);
    int bg   = *bg_p;
    int lbl  = pos ? gt_labels[b * NGT + g] : bg;
    out_labels[tid] = lbl;
    ((float4*)out_bboxes)[tid] = gtb4[pos ? g : 0];   // reference gathers gt 0 when unmatched
    float miou = pos ? iou_box(gtb4[g], pred4[tid]) : 0.0f;

    float4* srow = (float4*)(out_scores + (size_t)tid * NCLS);
    float4 z = {0.f, 0.f, 0.f, 0.f};
    #pragma unroll
    for (int j = 0; j < NCLS / 4; ++j) srow[j] = z;
    if (pos && lbl >= 0 && lbl < NCLS)
        out_scores[(size_t)tid * NCLS + lbl] = miou;
}

extern "C" void kernel_launch(void* const* d_in, const int* in_sizes, int n_in,
                              void* d_out, int out_size, void* d_ws, size_t ws_size,
                              hipStream_t stream) {
    const float* anchors   = (const float*)d_in[0];   // (8400,4)
    const int*   gt_labels = (const int*)  d_in[1];   // (32,128,1)
    const float* gt_bboxes = (const float*)d_in[2];   // (32,128,4)
    const float* pad_mask  = (const float*)d_in[3];   // (32,128,1)
    const float* pred      = (const float*)d_in[4];   // (32,8400,4)
    const int*   bg_p      = (const int*)  d_in[5];   // scalar

    const int nw = BATCH * NANCH;                     // 268,800
    char* wsp = (char*)d_ws;
    int* cnt = (int*)wsp;
    size_t off = (((size_t)nw * sizeof(int)) + 255) & ~(size_t)255;
    unsigned long long* best = (unsigned long long*)(wsp + off);
    off += (((size_t)nw * sizeof(unsigned long long)) + 255) & ~(size_t)255;
    float2* bf0 = (float2*)(wsp + off);
    off += (size_t)NANCH * sizeof(float2);
    float2* bf1 = (float2*)(wsp + off);

    init_prep_kernel<<<(nw + 255) / 256, 256, 0, stream>>>(anchors, cnt, best, bf0, bf1);

    // One (batch, 16-gt group) per 4-wave workgroup: 256 blocks x 128 threads.
    atss_phaseA_kernel<<<BATCH * (NGT / 16), NWAVES * 32, 0, stream>>>(
        anchors, gt_bboxes, pad_mask, bf0, bf1, cnt, best);

    int*   out_labels = (int*)d_out;
    float* out_bboxes = (float*)d_out + nw;
    float* out_scores = (float*)d_out + (size_t)nw * 5;
    atss_resolve_kernel<<<(nw + 255) / 256, 256, 0, stream>>>(
        anchors, gt_labels, gt_bboxes, pred, bg_p, cnt, best,
        out_labels, out_bboxes, out_scores);
}